// LRN_77695958384949
// MI455X (gfx1250) — compile-verified
//
#include <hip/hip_runtime.h>
#include <hip/hip_bf16.h>

#ifndef __has_builtin
#define __has_builtin(x) 0
#endif

#if __has_builtin(__builtin_amdgcn_global_load_async_to_lds_b128)
#define HAVE_ASYNC_LOAD 1
#else
#define HAVE_ASYNC_LOAD 0
#endif

#if __has_builtin(__builtin_amdgcn_global_store_async_from_lds_b128)
#define HAVE_ASYNC_STORE 1
#else
#define HAVE_ASYNC_STORE 0
#endif

#define C_CH            192
#define ROWS_PER_BLOCK  8
#define VEC_PER_ROW     (C_CH * 4 / 16)   // 48 float4 chunks per row
#define LRN_ALPHA_OVER_N 2.0e-5f          // 0.0001 / 5
#define LRN_K            1.0f

// The async-LDS builtins take int4-vector pointers qualified with explicit
// address spaces: arg0 = global (addrspace 1), arg1 = LDS (addrspace 3).
typedef int v4i __attribute__((ext_vector_type(4)));
typedef __attribute__((address_space(1))) v4i gv4i;
typedef __attribute__((address_space(3))) v4i lv4i;

__device__ __forceinline__ void wait_async_all() {
#if __has_builtin(__builtin_amdgcn_s_wait_asynccnt)
    __builtin_amdgcn_s_wait_asynccnt(0);
#elif (HAVE_ASYNC_LOAD || HAVE_ASYNC_STORE)
    asm volatile("s_wait_asynccnt 0" ::: "memory");
#endif
}

__global__ __launch_bounds__(256)
void lrn_buggy_window_kernel(const float* __restrict__ x,
                             float* __restrict__ out,
                             int nrows)
{
    // 8 rows staged per block: 8*192*4 = 6 KB input/output buffer,
    // 8*200*4 = 6.25 KB prefix-sum buffer (193 entries used, padded).
    __shared__ float xrow[ROWS_PER_BLOCK][C_CH];
    __shared__ float csum[ROWS_PER_BLOCK][200];

    const int lane = threadIdx.x & 31;
    const int wave = threadIdx.x >> 5;
    const int row  = blockIdx.x * ROWS_PER_BLOCK + wave;
    const bool active = row < nrows;   // wave-uniform

    const float4* gsrc = (const float4*)(x   + (size_t)row * C_CH);
    float4*       gdst = (float4*)(out + (size_t)row * C_CH);
    float4*       lrow = (float4*)&xrow[wave][0];

    // ---------------- Phase A: stage row into LDS (async b128) -------------
    if (active) {
#if HAVE_ASYNC_LOAD
        __builtin_amdgcn_global_load_async_to_lds_b128(
            (gv4i*)(gsrc + lane), (lv4i*)(lrow + lane), 0, 0);
        if (lane < VEC_PER_ROW - 32)
            __builtin_amdgcn_global_load_async_to_lds_b128(
                (gv4i*)(gsrc + lane + 32), (lv4i*)(lrow + lane + 32), 0, 0);
#else
        lrow[lane] = gsrc[lane];
        if (lane < VEC_PER_ROW - 32) lrow[lane + 32] = gsrc[lane + 32];
#endif
    }
    wait_async_all();
    __syncthreads();

    // ---------------- Phase B: squares + wave-wide exclusive prefix scan ---
    const int c0 = lane * 6;
    float xv[6]   = {0.f, 0.f, 0.f, 0.f, 0.f, 0.f};
    float pref[6] = {0.f, 0.f, 0.f, 0.f, 0.f, 0.f};
    float tot = 0.f;

    if (active) {
#pragma unroll
        for (int k = 0; k < 6; ++k) {
            float v = xrow[wave][c0 + k];
            xv[k] = v;
            tot += v * v;
            pref[k] = tot;
        }
    }

    // Kogge-Stone inclusive scan of per-lane totals across the wave32.
    float acc = tot;
#pragma unroll
    for (int d = 1; d < 32; d <<= 1) {
        float n = __int_as_float(
            __builtin_amdgcn_ds_bpermute((lane - d) << 2, __float_as_int(acc)));
        acc = (lane >= d) ? (acc + n) : acc;
    }
    const float excl = acc - tot;   // exclusive prefix of lane totals

    if (active) {
        if (lane == 0) csum[wave][0] = 0.f;
#pragma unroll
        for (int k = 0; k < 6; ++k)
            csum[wave][c0 + 1 + k] = excl + pref[k];  // csum[j] = sum_{i<j} x[i]^2
    }
    __syncthreads();

    // ---------------- Phase C: buggy-Keras window via prefix difference ----
    // window for channel c: [max(c-2,0), min(2c+3, C))
    if (active) {
#pragma unroll
        for (int k = 0; k < 6; ++k) {
            const int c    = c0 + k;
            const int head = (c >= 2) ? (c - 2) : 0;
            int       end  = 2 * c + 3;
            if (end > C_CH) end = C_CH;
            const float s  = csum[wave][end] - csum[wave][head];
            const float t  = fmaf(s, LRN_ALPHA_OVER_N, LRN_K);
            // x / t^0.75 == x * rsqrt(t * sqrt(t))
            xrow[wave][c] = xv[k] * rsqrtf(t * sqrtf(t));
        }
    }
    __syncthreads();

    // ---------------- Phase D: stream row back out (async b128 from LDS) ---
    if (active) {
#if HAVE_ASYNC_STORE
        __builtin_amdgcn_global_store_async_from_lds_b128(
            (gv4i*)(gdst + lane), (lv4i*)(lrow + lane), 0, 0);
        if (lane < VEC_PER_ROW - 32)
            __builtin_amdgcn_global_store_async_from_lds_b128(
                (gv4i*)(gdst + lane + 32), (lv4i*)(lrow + lane + 32), 0, 0);
#else
        gdst[lane] = lrow[lane];
        if (lane < VEC_PER_ROW - 32) gdst[lane + 32] = lrow[lane + 32];
#endif
    }
    wait_async_all();   // S_ENDPGM also does an implicit wait-idle
}

extern "C" void kernel_launch(void* const* d_in, const int* in_sizes, int n_in,
                              void* d_out, int out_size, void* d_ws, size_t ws_size,
                              hipStream_t stream) {
    (void)n_in; (void)out_size; (void)d_ws; (void)ws_size;
    const float* x  = (const float*)d_in[0];
    float*       out = (float*)d_out;

    const long long n     = (long long)in_sizes[0];
    const int       nrows = (int)(n / C_CH);              // 200704 for (64,56,56,192)
    const int       nblk  = (nrows + ROWS_PER_BLOCK - 1) / ROWS_PER_BLOCK;  // 25088

    lrn_buggy_window_kernel<<<nblk, 256, 0, stream>>>(x, out, nrows);
}